// GaussianImage_Cholesky_EA_8598524526682
// MI455X (gfx1250) — compile-verified
//
#include <hip/hip_runtime.h>
#include <math.h>

#define HH 256
#define WW 256
#define NPTS 1024
#define GGRP 2
#define RANKK 6
#define CCH 128
#define NPIX (HH * WW)
#define PSTRIDE 17      // cx,cy,ha,cb,hc, fx0,fy0,gw0, fx1,fy1,gw1, f0..f5
#define CHUNK 256

#define RENDER_ELEMS (CCH * NPIX)          // 8388608
#define ABUND_ELEMS  (NPIX * RANKK)        // 393216
#define E_ELEMS      (RANKK * CCH)         // 768

typedef __attribute__((ext_vector_type(2))) float v2f;
typedef __attribute__((ext_vector_type(8))) float v8f;

// ---------------- per-gaussian precompute: N=1024 ----------------
__global__ void k_precompute(const float* __restrict__ xyz,
                             const float* __restrict__ chol,
                             const float* __restrict__ fdc,
                             const float* __restrict__ gfreq,
                             const float* __restrict__ gwts,
                             const float* __restrict__ opac,
                             float* __restrict__ params) {
    int n = blockIdx.x * blockDim.x + threadIdx.x;
    if (n >= NPTS) return;
    float cx = (tanhf(xyz[n * 2 + 0]) * 0.5f + 0.5f) * (float)WW;
    float cy = (tanhf(xyz[n * 2 + 1]) * 0.5f + 0.5f) * (float)HH;
    float l1 = chol[n * 3 + 0] + 0.5f;
    float l2 = chol[n * 3 + 1];
    float l3 = chol[n * 3 + 2] + 0.5f;
    float a = l1 * l1;
    float b = l1 * l2;
    float c = l2 * l2 + l3 * l3;
    float inv = 1.0f / (a * c - b * b);
    float* p = params + n * PSTRIDE;
    p[0] = cx;
    p[1] = cy;
    p[2] = 0.5f * (c * inv);    // ha = 0.5*ca
    p[3] = -b * inv;            // cb
    p[4] = 0.5f * (a * inv);    // hc = 0.5*cc
#pragma unroll
    for (int g = 0; g < GGRP; ++g) {
        int idx = n * GGRP + g;
        p[5 + g * 3 + 0] = expf(gfreq[idx * 2 + 0]);            // fx
        p[5 + g * 3 + 1] = expf(gfreq[idx * 2 + 1]);            // fy
        p[5 + g * 3 + 2] = 1.0f / (1.0f + expf(-gwts[idx]));    // sigmoid
    }
    float op = opac[n];
#pragma unroll
    for (int r = 0; r < RANKK; ++r) {
        float x = fdc[n * RANKK + r];
        float sp = (x > 20.0f) ? x : log1pf(expf(x));           // softplus
        p[11 + r] = sp * op;
    }
}

// ---------------- E = clip(endmember + scale*tanh(U@V), 1e-6) ----------------
__global__ void k_endmember(const float* __restrict__ lu,
                            const float* __restrict__ lv,
                            const float* __restrict__ lsl,
                            const float* __restrict__ em,
                            float* __restrict__ Eout) {
    int i = blockIdx.x * blockDim.x + threadIdx.x;
    if (i >= RANKK * CCH) return;
    int r = i / CCH;
    int c = i - r * CCH;
    float d = lu[r * 2 + 0] * lv[c] + lu[r * 2 + 1] * lv[CCH + c];
    float scale = 0.1f * tanhf(lsl[0]);
    float e = em[i] + scale * tanhf(d);
    Eout[i] = fmaxf(e, 1e-6f);
}

// ---------------- main raster: abundance[p][r] ----------------
__global__ void __launch_bounds__(256)
k_raster(const float* __restrict__ params, float* __restrict__ abund) {
    __shared__ float sg[CHUNK * PSTRIDE];
    int tid = threadIdx.x;
    int p = blockIdx.x * 256 + tid;
    float px = (float)(p & (WW - 1)) + 0.5f;
    float py = (float)(p >> 8) + 0.5f;
    float acc[RANKK];
#pragma unroll
    for (int r = 0; r < RANKK; ++r) acc[r] = 0.0f;

    for (int cb = 0; cb < NPTS; cb += CHUNK) {
        for (int j = tid; j < CHUNK * PSTRIDE; j += 256)
            sg[j] = params[cb * PSTRIDE + j];
        __syncthreads();
        for (int i = 0; i < CHUNK; ++i) {
            const float* gp = &sg[i * PSTRIDE];
            float dx = px - gp[0];
            float dy = py - gp[1];
            float sigma = gp[2] * dx * dx + gp[3] * dx * dy + gp[4] * dy * dy;
            // whole-wave cull: contribution < e^-30 is below fp32 resolution
            if (__builtin_amdgcn_ballot_w32(sigma < 30.0f) == 0u) continue;
            float g = (sigma >= 0.0f) ? __expf(-sigma) : 0.0f;
            float mod = 1.0f + gp[7]  * __cosf(gp[5] * dx + gp[6] * dy)
                             + gp[10] * __cosf(gp[8] * dx + gp[9] * dy);
            float w = g * mod;
#pragma unroll
            for (int r = 0; r < RANKK; ++r) acc[r] += w * gp[11 + r];
        }
        __syncthreads();
    }
    float* o = abund + p * RANKK;
#pragma unroll
    for (int r = 0; r < RANKK; ++r) o[r] = fmaxf(acc[r], 0.0f);
}

// ---------------- render = (abundance @ E)^T via V_WMMA_F32_16X16X4_F32 ----
// D(16ch x 16px) = A(16x4: A[m][k]=E[k][cbase+m]) * B(4x16: B[k][n]=ab[pbase+n][k])
// K=6 -> two WMMAs (K 0..3, then K 4..5 zero-padded to 8).
__global__ void __launch_bounds__(256)
k_render(const float* __restrict__ abund, const float* __restrict__ E,
         float* __restrict__ render) {
    int lane = threadIdx.x & 31;
    int wave = threadIdx.x >> 5;      // 0..7 -> channel chunk
    int half = lane >> 4;             // 0: K={0,1}/{4,5}; 1: K={2,3}/{6,7}
    int m = lane & 15;
    int pbase = blockIdx.x * 16;
    int cbase = wave * 16;

    v2f a0, a1, b0, b1;
    int ch = cbase + m;
    const float* ar = abund + (pbase + m) * RANKK;
    if (half == 0) {
        a0.x = E[0 * CCH + ch]; a0.y = E[1 * CCH + ch];
        a1.x = E[4 * CCH + ch]; a1.y = E[5 * CCH + ch];
        b0.x = ar[0]; b0.y = ar[1];
        b1.x = ar[4]; b1.y = ar[5];
    } else {
        a0.x = E[2 * CCH + ch]; a0.y = E[3 * CCH + ch];
        a1.x = 0.0f; a1.y = 0.0f;     // K=6,7 pad
        b0.x = ar[2]; b0.y = ar[3];
        b1.x = 0.0f; b1.y = 0.0f;
    }

    v8f acc = {};
    acc = __builtin_amdgcn_wmma_f32_16x16x4_f32(false, a0, false, b0,
                                                (short)0, acc, false, false);
    acc = __builtin_amdgcn_wmma_f32_16x16x4_f32(false, a1, false, b1,
                                                (short)0, acc, false, false);

    // D: VGPR j, lanes 0-15 -> M=j; lanes 16-31 -> M=j+8; N = lane&15
    int pix = pbase + m;
#pragma unroll
    for (int j = 0; j < 8; ++j) {
        int c = cbase + j + (half ? 8 : 0);
        render[c * NPIX + pix] = acc[j];
    }
}

extern "C" void kernel_launch(void* const* d_in, const int* in_sizes, int n_in,
                              void* d_out, int out_size, void* d_ws, size_t ws_size,
                              hipStream_t stream) {
    (void)in_sizes; (void)n_in; (void)out_size; (void)ws_size;
    const float* xyz   = (const float*)d_in[0];
    const float* chol  = (const float*)d_in[1];
    const float* fdc   = (const float*)d_in[2];
    const float* gfreq = (const float*)d_in[3];
    const float* gwts  = (const float*)d_in[4];
    const float* lu    = (const float*)d_in[5];
    const float* lv    = (const float*)d_in[6];
    const float* lsl   = (const float*)d_in[7];
    const float* opac  = (const float*)d_in[8];
    const float* em    = (const float*)d_in[9];

    float* out    = (float*)d_out;
    float* render = out;                          // [C][H*W]
    float* abund  = out + RENDER_ELEMS;           // [H*W][RANK]
    float* Eout   = out + RENDER_ELEMS + ABUND_ELEMS;  // [RANK][C]
    float* params = (float*)d_ws;                 // [N][17]

    k_precompute<<<(NPTS + 255) / 256, 256, 0, stream>>>(xyz, chol, fdc, gfreq,
                                                         gwts, opac, params);
    k_endmember<<<(RANKK * CCH + 127) / 128, 128, 0, stream>>>(lu, lv, lsl, em, Eout);
    k_raster<<<NPIX / 256, 256, 0, stream>>>(params, abund);
    k_render<<<NPIX / 16, 256, 0, stream>>>(abund, Eout, render);
}